// HigherOrderScore_61718680043994
// MI455X (gfx1250) — compile-verified
//
#include <hip/hip_runtime.h>
#include <hip/hip_bf16.h>
#include <math.h>

// ---------------------------------------------------------------------------
// HigherOrderScore on MI455X (gfx1250, wave32, WMMA f32 16x16x4)
//
// All GEMMs use V_WMMA_F32_16X16X4_F32 (native CDNA5 fp32 matrix path) with
// LDS-staged tiles. Fragment layouts per CDNA5 ISA 7.12.2:
//   A (16x4 f32, v2f): lanes 0-15 -> M=lane, K={0,1}; lanes 16-31 -> K={2,3}
//   B (4x16 f32, v2f): VGPR0 -> rows K=0 (lanes 0-15) / K=2 (lanes 16-31),
//                      VGPR1 -> rows K=1 / K=3; N = lane%16
//   C/D (16x16 f32, v8f): elem e -> M = e + 8*(lane>=16), N = lane%16
// ---------------------------------------------------------------------------

typedef __attribute__((ext_vector_type(2))) float v2f;
typedef __attribute__((ext_vector_type(8))) float v8f;

#define NEGVAL   (-10000000000.0f)
#define NUM_MENT 1000
#define GI_DIM   768
#define KEFF     50
#define NPAIR    49950          /* 999 * 50 */
#define KDIM     2365           /* 3*768 + 20 + 1 + 2*20 */
#define HID      1000
#define PHI_DIM  61

static __device__ __forceinline__ v8f wmma_f32(v2f a, v2f b, v8f c) {
  return __builtin_amdgcn_wmma_f32_16x16x4_f32(false, a, false, b, (short)0, c,
                                               false, false);
}

// ---------------------------------------------------------------------------
// Kernel 1: tmp1[1000x768] = g[mids[m]] @ coarse_W^T
// tmp1[m][n] = sum_k G[mids[m]][k] * Wc[n][k]
// ---------------------------------------------------------------------------
__global__ __launch_bounds__(256)
void gemm_gWcT_kernel(const float* __restrict__ G, const float* __restrict__ Wc,
                      const int* __restrict__ Mids, float* __restrict__ T1) {
  __shared__ float As[32][33];
  __shared__ float Bs[32][65];
  int t = threadIdx.x;
  int lane = t & 31, wave = t >> 5;
  int wr = wave >> 2, wc = wave & 3;
  int half = lane >> 4, l16 = lane & 15;
  int m0 = blockIdx.y * 32, n0 = blockIdx.x * 64;
  v8f acc = {};
  for (int k0 = 0; k0 < GI_DIM; k0 += 32) {
    for (int i = 0; i < 4; ++i) {                 // 32x32 A tile
      int idx = t + i * 256;
      int kk = idx & 31, r = idx >> 5;
      int m = m0 + r;
      As[r][kk] = (m < NUM_MENT) ? G[(size_t)Mids[m] * GI_DIM + k0 + kk] : 0.0f;
    }
    for (int i = 0; i < 8; ++i) {                 // 32x64 B tile: Bs[k][n]=Wc[n][k]
      int idx = t + i * 256;
      int kk = idx & 31, n = idx >> 5;
      Bs[kk][n] = Wc[(size_t)(n0 + n) * GI_DIM + k0 + kk];
    }
    __syncthreads();
    for (int ks = 0; ks < 8; ++ks) {
      int ka = ks * 4 + half * 2;
      v2f a, b;
      a.x = As[wr * 16 + l16][ka];     a.y = As[wr * 16 + l16][ka + 1];
      b.x = Bs[ka][wc * 16 + l16];     b.y = Bs[ka + 1][wc * 16 + l16];
      acc = wmma_f32(a, b, acc);
    }
    __syncthreads();
  }
  int col = n0 + wc * 16 + l16;
  for (int e = 0; e < 8; ++e) {
    int row = m0 + wr * 16 + half * 8 + e;
    if (row < NUM_MENT) T1[(size_t)row * GI_DIM + col] = acc[e];
  }
}

// ---------------------------------------------------------------------------
// Kernel 2: coarse[1000x1000] = T1 @ g0^T + s_i + s_j + mask
// ---------------------------------------------------------------------------
__global__ __launch_bounds__(256)
void coarse_kernel(const float* __restrict__ T1, const float* __restrict__ G,
                   const float* __restrict__ Sc, const int* __restrict__ Mids,
                   float* __restrict__ Coarse) {
  __shared__ float As[32][33];
  __shared__ float Bs[32][65];
  int t = threadIdx.x;
  int lane = t & 31, wave = t >> 5;
  int wr = wave >> 2, wc = wave & 3;
  int half = lane >> 4, l16 = lane & 15;
  int m0 = blockIdx.y * 32, n0 = blockIdx.x * 64;
  v8f acc = {};
  for (int k0 = 0; k0 < GI_DIM; k0 += 32) {
    for (int i = 0; i < 4; ++i) {
      int idx = t + i * 256;
      int kk = idx & 31, r = idx >> 5;
      int m = m0 + r;
      As[r][kk] = (m < NUM_MENT) ? T1[(size_t)m * GI_DIM + k0 + kk] : 0.0f;
    }
    for (int i = 0; i < 8; ++i) {                 // Bs[k][n] = G[mids[n]][k]
      int idx = t + i * 256;
      int kk = idx & 31, n = idx >> 5;
      int gn = n0 + n;
      Bs[kk][n] = (gn < NUM_MENT) ? G[(size_t)Mids[gn] * GI_DIM + k0 + kk] : 0.0f;
    }
    __syncthreads();
    for (int ks = 0; ks < 8; ++ks) {
      int ka = ks * 4 + half * 2;
      v2f a, b;
      a.x = As[wr * 16 + l16][ka];     a.y = As[wr * 16 + l16][ka + 1];
      b.x = Bs[ka][wc * 16 + l16];     b.y = Bs[ka + 1][wc * 16 + l16];
      acc = wmma_f32(a, b, acc);
    }
    __syncthreads();
  }
  int col = n0 + wc * 16 + l16;
  float sn = (col < NUM_MENT) ? Sc[Mids[col]] : 0.0f;
  for (int e = 0; e < 8; ++e) {
    int row = m0 + wr * 16 + half * 8 + e;
    if (row < NUM_MENT && col < NUM_MENT) {
      float v = acc[e] + Sc[Mids[row]] + sn +
                ((row - col >= 1) ? 0.0f : NEGVAL);
      Coarse[(size_t)row * NUM_MENT + col] = v;
    }
  }
}

// ---------------------------------------------------------------------------
// Kernel 3: per-row top-50 (descending, ties -> lower index), like lax.top_k
// ---------------------------------------------------------------------------
__global__ __launch_bounds__(256)
void topk_kernel(const float* __restrict__ Coarse, float* __restrict__ BS,
                 int* __restrict__ BI) {
  __shared__ float vals[1024];
  __shared__ float rv[256];
  __shared__ int ri[256];
  int row = blockIdx.x;
  int t = threadIdx.x;
  for (int j = t; j < 1024; j += 256)
    vals[j] = (j < NUM_MENT) ? Coarse[(size_t)row * NUM_MENT + j] : -3.0e38f;
  __syncthreads();
  for (int it = 0; it < KEFF; ++it) {
    float best = -3.0e38f;
    int bidx = 0x7fffffff;
    for (int j = t; j < NUM_MENT; j += 256) {
      float v = vals[j];
      if (v > best || (v == best && j < bidx)) { best = v; bidx = j; }
    }
    rv[t] = best; ri[t] = bidx;
    __syncthreads();
    for (int s = 128; s > 0; s >>= 1) {
      if (t < s) {
        float ov = rv[t + s]; int oi = ri[t + s];
        if (ov > rv[t] || (ov == rv[t] && oi < ri[t])) { rv[t] = ov; ri[t] = oi; }
      }
      __syncthreads();
    }
    if (t == 0) {
      BS[row * KEFF + it] = rv[0];
      BI[row * KEFF + it] = ri[0];
      vals[ri[0]] = -3.0e38f;
    }
    __syncthreads();
  }
}

// ---------------------------------------------------------------------------
// Kernel 4: phi features (999 x 50 x 61)
// ---------------------------------------------------------------------------
__global__ __launch_bounds__(256)
void phi_kernel(const float* __restrict__ MT, const float* __restrict__ TE,
                const float* __restrict__ DistT, const int* __restrict__ BI,
                const int* __restrict__ Start, const int* __restrict__ End,
                float* __restrict__ Phi) {
  int p = blockIdx.x * blockDim.x + threadIdx.x;
  if (p >= NPAIR) return;
  int mrow = p / KEFF, a = p % KEFF;
  int i = mrow + 1;
  int j = BI[i * KEFF + a];
  int d = End[i] - Start[j];
  const int bins[11] = {1, 2, 3, 4, 8, 16, 32, 64, 128, 256, 384};
  int bidx = 0;
  for (int b = 0; b < 11; ++b) bidx += (d > bins[b]) ? 1 : 0;
  float* out = Phi + (size_t)p * PHI_DIM;
  const float* de = DistT + bidx * 20;
  const float* ti = MT + i * 20;
  const float* tj = MT + j * 20;
  float dot = 0.f, ni = 0.f, nj = 0.f;
  for (int c = 0; c < 20; ++c) {
    out[c] = de[c];
    float x = ti[c], y = tj[c];
    dot += x * y; ni += x * x; nj += y * y;
  }
  ni = fmaxf(sqrtf(ni), 1e-8f);
  nj = fmaxf(sqrtf(nj), 1e-8f);
  out[20] = dot / (ni * nj);
  const float* tei = TE + i * 20;
  const float* tej = TE + j * 20;
  for (int c = 0; c < 20; ++c) { out[21 + c] = tei[c]; out[41 + c] = tej[c]; }
}

// ---------------------------------------------------------------------------
// Kernel 5 (the big one, run twice): fused pair-MLP
//   coref[p] = valid ? leakyrelu(pairs_p @ W1 + b1) @ W2 + b2 + s_ij_c : NEG
// pairs_p (dim 2365) built on the fly from [I, J, I*J, phi].
// Block: 16 pair rows; 8 waves cover 256 hidden units/chunk (2 WMMA tiles each).
// ---------------------------------------------------------------------------
#define E_MT 16
#define E_NC 256
#define E_KC 32

__global__ __launch_bounds__(256)
void mlp_score_kernel(const float* __restrict__ G, const float* __restrict__ Phi,
                      const float* __restrict__ BS, const int* __restrict__ BI,
                      const int* __restrict__ Mids, const float* __restrict__ W1,
                      const float* __restrict__ B1, const float* __restrict__ W2,
                      const float* __restrict__ B2, float* __restrict__ Coref) {
  __shared__ float As[E_MT][E_KC + 1];
  __shared__ float Bs[E_KC][E_NC + 1];
  __shared__ float sacc[E_MT];
  __shared__ int mg[E_MT], ag[E_MT];

  int t = threadIdx.x;
  int lane = t & 31, wave = t >> 5;
  int half = lane >> 4, l16 = lane & 15;
  int p0 = blockIdx.x * E_MT;

  if (t < E_MT) {
    int p = p0 + t;
    int mgid = 0, agid = 0;
    if (p < NPAIR) {
      int mrow = p / KEFF, a = p % KEFF;
      mgid = Mids[mrow + 1];
      agid = Mids[BI[(mrow + 1) * KEFF + a]];
    }
    mg[t] = mgid; ag[t] = agid;
    sacc[t] = 0.0f;
  }
  __syncthreads();

  const int KCHUNKS = (KDIM + E_KC - 1) / E_KC;  // 74
  for (int nc = 0; nc < 4; ++nc) {
    int n0 = nc * E_NC;
    v8f acc0 = {}, acc1 = {};
    for (int kc = 0; kc < KCHUNKS; ++kc) {
      int k0 = kc * E_KC;
      // A tile 16x32 (512 elems, 2/thread): gather/compute pair features
      {
        int idx = t;
        for (int i = 0; i < 2; ++i, idx += 256) {
          int kk = idx & 31, r = idx >> 5;
          int gk = k0 + kk;
          int p = p0 + r;
          float v = 0.0f;
          if (p < NPAIR && gk < KDIM) {
            if (gk < 768) {
              v = G[(size_t)mg[r] * GI_DIM + gk];
            } else if (gk < 1536) {
              v = G[(size_t)ag[r] * GI_DIM + (gk - 768)];
            } else if (gk < 2304) {
              int kf = gk - 1536;
              v = G[(size_t)mg[r] * GI_DIM + kf] * G[(size_t)ag[r] * GI_DIM + kf];
            } else {
              v = Phi[(size_t)p * PHI_DIM + (gk - 2304)];
            }
          }
          As[r][kk] = v;
        }
      }
      // B tile 32x256 (8192 elems, 32/thread) from W1 (K x HID row-major)
      {
        int idx = t;
        for (int i = 0; i < 32; ++i, idx += 256) {
          int n = idx & (E_NC - 1), kk = idx >> 8;
          int gk = k0 + kk, gn = n0 + n;
          Bs[kk][n] = (gk < KDIM && gn < HID) ? W1[(size_t)gk * HID + gn] : 0.0f;
        }
      }
      __syncthreads();
      for (int ks = 0; ks < E_KC / 4; ++ks) {
        int ka = ks * 4 + half * 2;
        v2f a, b0, b1v;
        a.x = As[l16][ka];                      a.y = As[l16][ka + 1];
        b0.x = Bs[ka][wave * 32 + l16];         b0.y = Bs[ka + 1][wave * 32 + l16];
        b1v.x = Bs[ka][wave * 32 + 16 + l16];   b1v.y = Bs[ka + 1][wave * 32 + 16 + l16];
        acc0 = wmma_f32(a, b0, acc0);
        acc1 = wmma_f32(a, b1v, acc1);
      }
      __syncthreads();
    }
    // Epilogue: h = leakyrelu(acc + b1); sacc[row] += h * W2[n]
    int na = n0 + wave * 32 + l16;
    int nb = na + 16;
    float b1a = (na < HID) ? B1[na] : 0.0f;
    float b1b = (nb < HID) ? B1[nb] : 0.0f;
    float w2a = (na < HID) ? W2[na] : 0.0f;
    float w2b = (nb < HID) ? W2[nb] : 0.0f;
    for (int e = 0; e < 8; ++e) {
      int row = half * 8 + e;
      float ha = acc0[e] + b1a; ha = (ha >= 0.0f) ? ha : 0.01f * ha;
      float hb = acc1[e] + b1b; hb = (hb >= 0.0f) ? hb : 0.01f * hb;
      atomicAdd(&sacc[row], ha * w2a + hb * w2b);
    }
    __syncthreads();
  }
  if (t < E_MT) {
    int p = p0 + t;
    if (p < NPAIR) {
      int mrow = p / KEFF, a = p % KEFF;
      int lim = (mrow + 1 < KEFF) ? (mrow + 1) : KEFF;
      float val = NEGVAL;
      if (a < lim) val = sacc[t] + B2[0] + BS[(mrow + 1) * KEFF + a];
      Coref[p] = val;
    }
  }
}

// ---------------------------------------------------------------------------
// Kernel 6: softmax over [0, coref_row] and attention-weighted antecedent a_n
// ---------------------------------------------------------------------------
__global__ __launch_bounds__(256)
void attend_kernel(const float* __restrict__ G, const float* __restrict__ Coref,
                   const int* __restrict__ BI, const int* __restrict__ Mids,
                   float* __restrict__ AN) {
  __shared__ float w[KEFF + 1];
  __shared__ int jr[KEFF];
  __shared__ int irow;
  int row = blockIdx.x;  // 0..998
  int t = threadIdx.x;
  if (t == 0) { w[0] = 0.0f; irow = Mids[row + 1]; }
  if (t < KEFF) {
    w[t + 1] = Coref[row * KEFF + t];
    jr[t] = Mids[BI[(row + 1) * KEFF + t]];
  }
  __syncthreads();
  if (t == 0) {
    float mx = w[0];
    for (int i = 1; i <= KEFF; ++i) mx = fmaxf(mx, w[i]);
    float s = 0.0f;
    for (int i = 0; i <= KEFF; ++i) { w[i] = expf(w[i] - mx); s += w[i]; }
    float inv = 1.0f / s;
    for (int i = 0; i <= KEFF; ++i) w[i] *= inv;
  }
  __syncthreads();
  int ir = irow;
  for (int d = t; d < GI_DIM; d += 256) {
    float acc = w[0] * G[(size_t)ir * GI_DIM + d];
    for (int a = 0; a < KEFF; ++a)
      acc += w[a + 1] * G[(size_t)jr[a] * GI_DIM + d];
    AN[(size_t)row * GI_DIM + d] = acc;
  }
}

// ---------------------------------------------------------------------------
// Kernel 7: gated update  g'[mids[m+1]] = f*I + (1-f)*a_n,
//           f = sigmoid([I, a_n] @ W_f + b_f).  WMMA GEMM 999x768, K=1536.
// ---------------------------------------------------------------------------
__global__ __launch_bounds__(256)
void update_g_kernel(const float* __restrict__ G, const float* __restrict__ AN,
                     const float* __restrict__ Wf, const float* __restrict__ bf,
                     const int* __restrict__ Mids, float* __restrict__ Gout) {
  __shared__ float As[32][33];
  __shared__ float Bs[32][65];
  int t = threadIdx.x;
  int lane = t & 31, wave = t >> 5;
  int wr = wave >> 2, wc = wave & 3;
  int half = lane >> 4, l16 = lane & 15;
  int m0 = blockIdx.y * 32, n0 = blockIdx.x * 64;
  v8f acc = {};
  for (int k0 = 0; k0 < 2 * GI_DIM; k0 += 32) {
    for (int i = 0; i < 4; ++i) {
      int idx = t + i * 256;
      int kk = idx & 31, r = idx >> 5;
      int m = m0 + r;
      float v = 0.0f;
      if (m < NUM_MENT - 1) {
        int gk = k0 + kk;
        if (gk < GI_DIM) v = G[(size_t)Mids[m + 1] * GI_DIM + gk];
        else             v = AN[(size_t)m * GI_DIM + (gk - GI_DIM)];
      }
      As[r][kk] = v;
    }
    for (int i = 0; i < 8; ++i) {                 // Bs[k][n] = Wf[k][n] row-major
      int idx = t + i * 256;
      int n = idx & 63, kk = idx >> 6;
      Bs[kk][n] = Wf[(size_t)(k0 + kk) * GI_DIM + n0 + n];
    }
    __syncthreads();
    for (int ks = 0; ks < 8; ++ks) {
      int ka = ks * 4 + half * 2;
      v2f a, b;
      a.x = As[wr * 16 + l16][ka];     a.y = As[wr * 16 + l16][ka + 1];
      b.x = Bs[ka][wc * 16 + l16];     b.y = Bs[ka + 1][wc * 16 + l16];
      acc = wmma_f32(a, b, acc);
    }
    __syncthreads();
  }
  int col = n0 + wc * 16 + l16;
  for (int e = 0; e < 8; ++e) {
    int m = m0 + wr * 16 + half * 8 + e;
    if (m < NUM_MENT - 1) {
      int gi = Mids[m + 1];
      float I = G[(size_t)gi * GI_DIM + col];
      float an = AN[(size_t)m * GI_DIM + col];
      float fv = 1.0f / (1.0f + expf(-(acc[e] + bf[col])));
      Gout[(size_t)gi * GI_DIM + col] = fv * I + (1.0f - fv) * an;
    }
  }
}

// ---------------------------------------------------------------------------
// Kernel 8: assemble final (1000, 51) scores
// ---------------------------------------------------------------------------
__global__ __launch_bounds__(256)
void assemble_kernel(const float* __restrict__ Coref, float* __restrict__ Out) {
  int idx = blockIdx.x * blockDim.x + threadIdx.x;
  if (idx >= NUM_MENT * (KEFF + 1)) return;
  int r = idx / (KEFF + 1), c = idx % (KEFF + 1);
  float v;
  if (c == 0)      v = 0.0f;
  else if (r == 0) v = (c == 1) ? 0.0f : NEGVAL;
  else             v = Coref[(r - 1) * KEFF + (c - 1)];
  Out[idx] = v;
}

// ---------------------------------------------------------------------------
extern "C" void kernel_launch(void* const* d_in, const int* in_sizes, int n_in,
                              void* d_out, int out_size, void* d_ws, size_t ws_size,
                              hipStream_t stream) {
  const float* G   = (const float*)d_in[0];   // g_i (5000,768)
  const float* Sc  = (const float*)d_in[1];   // mention_scores (5000,1)
  const float* MT  = (const float*)d_in[2];   // mention_types (1000,20)
  const float* TE  = (const float*)d_in[3];   // type_embeds (1000,20)
  const float* Wc  = (const float*)d_in[4];   // coarse_W (768,768)
  const float* DT  = (const float*)d_in[5];   // dist_table (12,20)
  const float* W1  = (const float*)d_in[6];   // (2365,1000)
  const float* B1  = (const float*)d_in[7];   // (1000,)
  const float* W2  = (const float*)d_in[8];   // (1000,1)
  const float* B2  = (const float*)d_in[9];   // (1,)
  const float* Wf  = (const float*)d_in[10];  // (1536,768)
  const float* bf  = (const float*)d_in[11];  // (768,)
  const int* Mids  = (const int*)d_in[12];    // (1000,)
  const int* Start = (const int*)d_in[13];    // (1000,)
  const int* End   = (const int*)d_in[14];    // (1000,)

  float* ws     = (float*)d_ws;
  float* tmp1   = ws;                         // 768,000 f
  float* coarse = tmp1 + 768000;              // 1,000,000 f
  float* bscore = coarse + 1000000;           // 50,000 f
  int*   bidx   = (int*)(bscore + 50000);     // 50,000 i32
  float* phi    = (float*)(bidx + 50000);     // 3,046,950 f
  float* coref  = phi + 3046950;              // 49,950 f
  float* an     = coref + 49950;              // 767,232 f
  float* g1     = an + 767232;                // 768,000 f   (total ~26 MB)

  dim3 blk(256);

  // 1) tmp1 = g0 @ Wc^T
  gemm_gWcT_kernel<<<dim3(GI_DIM / 64, 32), blk, 0, stream>>>(G, Wc, Mids, tmp1);
  // 2) coarse scores
  coarse_kernel<<<dim3(16, 32), blk, 0, stream>>>(tmp1, G, Sc, Mids, coarse);
  // 3) top-50 per row
  topk_kernel<<<dim3(NUM_MENT), blk, 0, stream>>>(coarse, bscore, bidx);
  // 4) phi features
  phi_kernel<<<dim3((NPAIR + 255) / 256), blk, 0, stream>>>(MT, TE, DT, bidx,
                                                            Start, End, phi);
  // 5) step 0: fused pair-MLP on original g
  mlp_score_kernel<<<dim3((NPAIR + E_MT - 1) / E_MT), blk, 0, stream>>>(
      G, phi, bscore, bidx, Mids, W1, B1, W2, B2, coref);
  // 6) softmax attention -> a_n
  attend_kernel<<<dim3(NUM_MENT - 1), blk, 0, stream>>>(G, coref, bidx, Mids, an);
  // 7) seed updated-g buffer (rows 0..999), then gated update of rows 1..999
  hipMemcpyAsync(g1, G, (size_t)NUM_MENT * GI_DIM * sizeof(float),
                 hipMemcpyDeviceToDevice, stream);
  update_g_kernel<<<dim3(GI_DIM / 64, 32), blk, 0, stream>>>(G, an, Wf, bf, Mids, g1);
  // 8) step 1: fused pair-MLP on updated g
  mlp_score_kernel<<<dim3((NPAIR + E_MT - 1) / E_MT), blk, 0, stream>>>(
      g1, phi, bscore, bidx, Mids, W1, B1, W2, B2, coref);
  // 9) assemble output (1000, 51)
  assemble_kernel<<<dim3((NUM_MENT * (KEFF + 1) + 255) / 256), blk, 0, stream>>>(
      coref, (float*)d_out);
}